// OscBank_70205535421335
// MI455X (gfx1250) — compile-verified
//
#include <hip/hip_runtime.h>
#include <math.h>

typedef __attribute__((ext_vector_type(2))) float v2f;
typedef __attribute__((ext_vector_type(8))) float v8f;

#define N_OSC     1024
#define N_SAMPLES 16384
#define NB        128          // partial-sum blocks (rows of P)
#define ROWS_PER_BLOCK 8       // 8 waves of 32 = 256 threads, one wave per oscillator row

// ---------------------------------------------------------------------------
// Kernel 1: per-row clamp + cumsum (wave32 shuffle scan) + sin*amp, with an
// in-block LDS reduction over the 8 rows this block owns.  Writes one partial
// row P[blockIdx][t] (or atomically accumulates into out[] in fallback mode).
// ---------------------------------------------------------------------------
__global__ __launch_bounds__(256)
void oscbank_scan(const float* __restrict__ freq_in,
                  const float* __restrict__ amp_in,
                  float* __restrict__ P,          // [NB][N_SAMPLES] partials
                  float* __restrict__ out_atomic, // fallback target
                  int use_atomic)
{
    __shared__ float lds[ROWS_PER_BLOCK * 128];

    const int wave = threadIdx.x >> 5;
    const int lane = threadIdx.x & 31;
    const int row  = blockIdx.x * ROWS_PER_BLOCK + wave;

    const float* __restrict__ fr = freq_in + (size_t)row * N_SAMPLES;
    const float* __restrict__ ar = amp_in  + (size_t)row * N_SAMPLES;

    const float TWO_PI = 6.28318530717958647692f; // rounds to fp32 2*pi, as in the reference

    float carry = 0.0f;  // running cumsum, wave-uniform

    for (int i = 0; i < N_SAMPLES / 128; ++i) {
        const int t0 = i * 128;
        const int tl = lane * 4;

        float4 f = *(const float4*)(fr + t0 + tl);
        float4 a = *(const float4*)(ar + t0 + tl);

        // clamp to [0,1]
        f.x = fminf(fmaxf(f.x, 0.0f), 1.0f);
        f.y = fminf(fmaxf(f.y, 0.0f), 1.0f);
        f.z = fminf(fmaxf(f.z, 0.0f), 1.0f);
        f.w = fminf(fmaxf(f.w, 0.0f), 1.0f);
        a.x = fminf(fmaxf(a.x, 0.0f), 1.0f);
        a.y = fminf(fmaxf(a.y, 0.0f), 1.0f);
        a.z = fminf(fmaxf(a.z, 0.0f), 1.0f);
        a.w = fminf(fmaxf(a.w, 0.0f), 1.0f);

        // lane-local inclusive prefix of 4
        float p0 = f.x;
        float p1 = p0 + f.y;
        float p2 = p1 + f.z;
        float p3 = p2 + f.w;

        // wave32 inclusive scan of the lane sums
        float v = p3;
        #pragma unroll
        for (int d = 1; d < 32; d <<= 1) {
            float n = __shfl_up(v, d, 32);
            if (lane >= d) v += n;
        }
        const float excl = v - p3;           // exclusive scan for this lane
        const float base = carry + excl;
        carry += __shfl(v, 31, 32);          // add wave total (broadcast, stays uniform)

        // oscillate + amplitude
        float s0 = sinf(TWO_PI * (base + p0)) * a.x;
        float s1 = sinf(TWO_PI * (base + p1)) * a.y;
        float s2 = sinf(TWO_PI * (base + p2)) * a.z;
        float s3 = sinf(TWO_PI * (base + p3)) * a.w;

        float* l = lds + wave * 128 + tl;
        l[0] = s0; l[1] = s1; l[2] = s2; l[3] = s3;
        __syncthreads();

        if (threadIdx.x < 128) {
            float s = 0.0f;
            #pragma unroll
            for (int w = 0; w < ROWS_PER_BLOCK; ++w)
                s += lds[w * 128 + threadIdx.x];
            const int t = t0 + threadIdx.x;
            if (use_atomic)
                atomicAdd(out_atomic + t, s);
            else
                P[(size_t)blockIdx.x * N_SAMPLES + t] = s;
        }
        __syncthreads();
    }
}

// ---------------------------------------------------------------------------
// Kernel 2: reduce P[128][t] -> out[t] using V_WMMA_F32_16X16X4_F32.
// A = all-ones 16x4 (all-ones is invariant under any VGPR striping), B holds
// 64 partials of a single t per tile.  D[m][n] = sum_k B[k][n], so summing all
// 256 D slots gives 16 * (sum of all B values) independent of the exact
// (lane,VGPR)->(k,n) mapping.  Two chained WMMAs cover all 128 partials.
// One wave per output sample t; 8 waves per block.
// ---------------------------------------------------------------------------
__global__ __launch_bounds__(256)
void reduce_wmma(const float* __restrict__ P, float* __restrict__ out)
{
    const int wave = threadIdx.x >> 5;
    const int lane = threadIdx.x & 31;
    const int t    = blockIdx.x * 8 + wave;

    v2f a_ones; a_ones.x = 1.0f; a_ones.y = 1.0f;

    v2f b0, b1;
    b0.x = P[(size_t)(lane      ) * N_SAMPLES + t];
    b0.y = P[(size_t)(lane + 32 ) * N_SAMPLES + t];
    b1.x = P[(size_t)(lane + 64 ) * N_SAMPLES + t];
    b1.y = P[(size_t)(lane + 96 ) * N_SAMPLES + t];

    v8f c = {};
    // 8 args: (neg_a, A, neg_b, B, c_mod, C, reuse_a, reuse_b)
    c = __builtin_amdgcn_wmma_f32_16x16x4_f32(false, a_ones, false, b0,
                                              (short)0, c, false, false);
    c = __builtin_amdgcn_wmma_f32_16x16x4_f32(false, a_ones, false, b1,
                                              (short)0, c, false, false);

    float s = c[0] + c[1] + c[2] + c[3] + c[4] + c[5] + c[6] + c[7];
    #pragma unroll
    for (int d = 16; d >= 1; d >>= 1)
        s += __shfl_xor(s, d, 32);

    if (lane == 0)
        out[t] = s * (1.0f / 16.0f);
}

__global__ void zero_out(float* __restrict__ out, int n)
{
    int i = blockIdx.x * 256 + threadIdx.x;
    if (i < n) out[i] = 0.0f;
}

// ---------------------------------------------------------------------------
extern "C" void kernel_launch(void* const* d_in, const int* in_sizes, int n_in,
                              void* d_out, int out_size, void* d_ws, size_t ws_size,
                              hipStream_t stream)
{
    // inputs: d_in[0] = x (unused), d_in[1] = osc_samples, d_in[2] = amplitudes
    const float* freq = (const float*)d_in[1];
    const float* amp  = (const float*)d_in[2];
    float* out = (float*)d_out;

    const size_t needed = (size_t)NB * N_SAMPLES * sizeof(float); // 8 MB

    if (ws_size >= needed) {
        float* P = (float*)d_ws;
        oscbank_scan<<<NB, 256, 0, stream>>>(freq, amp, P, nullptr, 0);
        reduce_wmma<<<N_SAMPLES / 8, 256, 0, stream>>>(P, out);
    } else {
        // fallback: accumulate directly into zeroed output
        zero_out<<<(N_SAMPLES + 255) / 256, 256, 0, stream>>>(out, N_SAMPLES);
        oscbank_scan<<<NB, 256, 0, stream>>>(freq, amp, nullptr, out, 1);
    }
}